// HierarchicalLocationEmbedding_25623774888460
// MI455X (gfx1250) — compile-verified
//
#include <hip/hip_runtime.h>
#include <math.h>

// ---------------------------------------------------------------------------
// HierarchicalLocationEmbedding for MI455X (gfx1250, wave32).
//
// Memory-bound problem (~50 MB moved, ~164 MFLOP) -> two kernels:
//   K1: gather fine rows (float4), gather+softmax 50 logits per token,
//       weights zero-padded to K=64 into workspace.
//   K2: (12800 x 52pad) @ (52pad x 128) GEMM with V_WMMA_F32_16X16X4_F32.
//       B staged in LDS in K-pair-interleaved layout so every B fragment is
//       one aligned ds_load_b64 (no register shuffles before WMMA).
// ---------------------------------------------------------------------------

typedef __attribute__((ext_vector_type(2))) float v2f;
typedef __attribute__((ext_vector_type(8))) float v8f;

#define NUM_LOC  100000
#define EMBED    256
#define NC       50
#define NTOK     (64 * 200)   // 12800 tokens
#define DOUT     384          // 256 fine + 128 coarse
#define KPAD     64           // padded weight row (floats) -> 256B rows
#define KSTEPS   13           // K = 52 = 13 x 4
#define KPAIRS   26           // 52 / 2 interleaved K-pairs
#define NDIM     128          // coarse dim

// ---------------------------------------------------------------------------
// Kernel 1: one wave per token.
//  - copy fine row (256 f32 = 64 float4, 2 per lane) directly into out
//  - gather 50 logits (lane holds clusters lane and lane+32), softmax via
//    wave32 shfl_xor reductions
//  - store weights zero-padded to KPAD into workspace
// ---------------------------------------------------------------------------
__global__ __launch_bounds__(256) void hle_gather_softmax_kernel(
    const int*   __restrict__ loc_ids,
    const float* __restrict__ loc_table,
    const float* __restrict__ W_lc,
    const float* __restrict__ b_lc,
    float*       __restrict__ out,
    float*       __restrict__ wts)
{
    const int wave = blockIdx.x * (blockDim.x >> 5) + (threadIdx.x >> 5);
    const int lane = threadIdx.x & 31;
    if (wave >= NTOK) return;

    const int id = loc_ids[wave];

    // ---- fine: 1 KB row copy, fully coalesced 16B per lane x 2 ----
    const float4* __restrict__ src = (const float4*)(loc_table + (size_t)id * EMBED);
    float4*       __restrict__ dst = (float4*)(out + (size_t)wave * DOUT);
    dst[lane]      = src[lane];
    dst[lane + 32] = src[lane + 32];

    // ---- logits: column gather of W_lc + bias; softmax over 50 ----
    const int c0 = lane;
    const int c1 = lane + 32;
    float l0 = (c0 < NC) ? (W_lc[(size_t)c0 * NUM_LOC + id] + b_lc[c0]) : -INFINITY;
    float l1 = (c1 < NC) ? (W_lc[(size_t)c1 * NUM_LOC + id] + b_lc[c1]) : -INFINITY;

    float m = fmaxf(l0, l1);
    #pragma unroll
    for (int off = 16; off > 0; off >>= 1)
        m = fmaxf(m, __shfl_xor(m, off, 32));

    float e0 = (c0 < NC) ? expf(l0 - m) : 0.0f;
    float e1 = (c1 < NC) ? expf(l1 - m) : 0.0f;

    float s = e0 + e1;
    #pragma unroll
    for (int off = 16; off > 0; off >>= 1)
        s += __shfl_xor(s, off, 32);

    const float inv = 1.0f / s;
    float* __restrict__ wrow = wts + (size_t)wave * KPAD;
    wrow[lane]      = e0 * inv;   // zero for lanes >= NC by construction
    wrow[lane + 32] = e1 * inv;   // zero beyond cluster 49 -> K padding is free
}

// ---------------------------------------------------------------------------
// Kernel 2: coarse = weights @ cluster_table via V_WMMA_F32_16X16X4_F32.
// One wave per 16-token M-tile; 8 N-tiles of 16; K = 52 (13 x K4 steps).
//
// Fragment layouts (ISA 7.12.2, 32-bit, wave32):
//   A 16x4 : lanes 0-15 -> M=lane,    {V0,V1} = K {4k+0, 4k+1}
//            lanes16-31 -> M=lane-16, {V0,V1} = K {4k+2, 4k+3}
//   B 4x16 : lanes 0-15 -> N=lane,    {V0,V1} = K {4k+0, 4k+1}
//            lanes16-31 -> N=lane-16, {V0,V1} = K {4k+2, 4k+3}
//   C 16x16: VGPR j -> row j (lanes 0-15) / row j+8 (lanes 16-31), N = lane&15
//
// LDS holds cluster_table K-pair interleaved: ctab2[p][n] = {CT[2p][n],
// CT[2p+1][n]} so a B fragment (K pair 2k+half, column n) is one ds_load_b64.
// ---------------------------------------------------------------------------
__global__ __launch_bounds__(256) void hle_coarse_wmma_kernel(
    const float* __restrict__ wts,
    const float* __restrict__ cluster_table,
    float*       __restrict__ out)
{
    __shared__ v2f ctab2[KPAIRS * NDIM];   // 26.6 KB of 320 KB/WGP

    // Stage cluster_table into LDS, interleaving K-pairs; zero rows 50,51.
    for (int i = threadIdx.x; i < KPAIRS * NDIM; i += blockDim.x) {
        const int p = i >> 7;          // K-pair index: rows 2p, 2p+1
        const int n = i & (NDIM - 1);  // column
        v2f v;
        v.x = (2 * p     < NC) ? cluster_table[(2 * p)     * NDIM + n] : 0.0f;
        v.y = (2 * p + 1 < NC) ? cluster_table[(2 * p + 1) * NDIM + n] : 0.0f;
        ctab2[i] = v;
    }
    __syncthreads();

    const int wave = blockIdx.x * (blockDim.x >> 5) + (threadIdx.x >> 5);
    const int lane = threadIdx.x & 31;
    const int m0   = wave * 16;
    if (m0 >= NTOK) return;

    const int half = lane >> 4;     // 0: lanes 0-15, 1: lanes 16-31
    const int l15  = lane & 15;

    v8f acc[8] = {};                // 8 N-tiles of 16x16 f32 accumulators

    // A-fragment base for this lane: row (m0+l15), K offset half*2.
    // KPAD=64 -> 256B rows; (4k + 2*half)*4 bytes is always 8B aligned.
    const float* __restrict__ arow = wts + (size_t)(m0 + l15) * KPAD + (half << 1);

    // B-fragment base in LDS for this lane: pair (2k+half), column l15.
    const v2f* __restrict__ brow = ctab2 + half * NDIM + l15;

    #pragma unroll
    for (int k = 0; k < KSTEPS; ++k) {
        const v2f a = *(const v2f*)(arow + 4 * k);       // K = 4k+{0,1}/{2,3}
        const v2f* __restrict__ bp = brow + 2 * k * NDIM;
        #pragma unroll
        for (int nt = 0; nt < 8; ++nt) {
            const v2f b = bp[nt * 16];                   // single ds_load_b64
            acc[nt] = __builtin_amdgcn_wmma_f32_16x16x4_f32(
                false, a, false, b, (short)0, acc[nt], false, false);
        }
    }

    // Scatter C tiles into the coarse slice of out.
    #pragma unroll
    for (int nt = 0; nt < 8; ++nt) {
        const int col = 256 + nt * 16 + l15;
        #pragma unroll
        for (int j = 0; j < 8; ++j) {
            const int row = m0 + j + (half << 3);
            out[(size_t)row * DOUT + col] = acc[nt][j];
        }
    }
}

// ---------------------------------------------------------------------------
// launch
// ---------------------------------------------------------------------------
extern "C" void kernel_launch(void* const* d_in, const int* in_sizes, int n_in,
                              void* d_out, int out_size, void* d_ws, size_t ws_size,
                              hipStream_t stream) {
    const int*   loc_ids       = (const int*)  d_in[0];  // (64, 200)
    const float* loc_table     = (const float*)d_in[1];  // (100000, 256)
    const float* cluster_table = (const float*)d_in[2];  // (50, 128)
    const float* W_lc          = (const float*)d_in[3];  // (50, 100000)
    const float* b_lc          = (const float*)d_in[4];  // (50,)
    float*       out           = (float*)d_out;          // (12800, 384)
    float*       wts           = (float*)d_ws;           // (12800, 64) padded weights

    // K1: 12800 waves, 8 waves (256 threads) per block -> 1600 blocks.
    hle_gather_softmax_kernel<<<NTOK / 8, 256, 0, stream>>>(
        loc_ids, loc_table, W_lc, b_lc, out, wts);

    // K2: 800 M-tile waves, 8 per block -> 100 blocks.
    hle_coarse_wmma_kernel<<<(NTOK / 16) / 8, 256, 0, stream>>>(
        wts, cluster_table, out);
}